// MultiHeadGraphAttention_75874892251862
// MI455X (gfx1250) — compile-verified
//
#include <hip/hip_runtime.h>
#include <hip/hip_bf16.h>
#include <math.h>

#define N_NODES 50000
#define N_EDGES 800000
#define DIM     128
#define HEADS   8
#define UNITS   32
#define FEAT    (HEADS*UNITS)   // 256
#define SLOPE   0.2f
#define EPS_F   1e-7f

typedef __attribute__((ext_vector_type(2))) float v2f;
typedef __attribute__((ext_vector_type(8))) float v8f;

// ---- order-preserving float<->uint encoding for atomic segment-max ----
__device__ __forceinline__ unsigned enc_f(float f) {
    unsigned b = __float_as_uint(f);
    return (b & 0x80000000u) ? ~b : (b | 0x80000000u);
}
__device__ __forceinline__ float dec_f(unsigned k) {
    unsigned b = (k & 0x80000000u) ? (k & 0x7fffffffu) : ~k;
    return __uint_as_float(b);
}

// ---- 0: init accumulators (d_out is poisoned by harness; we own zeroing) ----
__global__ void gat_init(float* __restrict__ out, unsigned* __restrict__ smax,
                         float* __restrict__ den) {
    int i = blockIdx.x * blockDim.x + threadIdx.x;
    if (i < N_NODES * FEAT)  out[i] = 0.0f;
    if (i < N_NODES * HEADS) { smax[i] = 0u; den[i] = 0.0f; }
}

// ---- 1: xp = x @ kernel via V_WMMA_F32_16X16X4_F32 ----
// One wave computes one 16x16 tile of xp. N=50000 is a multiple of 16, so
// every wave is full and EXEC is all-ones (WMMA requirement).
// A (16x4 f32) lane layout: lanes 0-15 -> M=l, K={k,k+1}; lanes 16-31 -> K={k+2,k+3}.
// B (4x16 f32) lane layout: lanes 0-15 -> N=l, same K split.
__global__ __launch_bounds__(256) void gat_gemm_wmma(
        const float* __restrict__ x, const float* __restrict__ w,
        float* __restrict__ xp) {
    const int wave = (blockIdx.x * blockDim.x + threadIdx.x) >> 5;
    const int lane = threadIdx.x & 31;
    const int tile_m = wave >> 4;          // 3125 row tiles
    const int tile_n = wave & 15;          // 16 col tiles
    const int m0 = tile_m * 16, n0 = tile_n * 16;
    const int half = lane >> 4;            // 0 or 1 (K sub-pair)
    const int l    = lane & 15;

    const float* arow = x + (size_t)(m0 + l) * DIM + 2 * half;   // A: M = l
    const float* bcol = w + (size_t)(2 * half) * FEAT + (n0 + l);// B: N = l

    v8f c = {};
    #pragma unroll 8
    for (int k = 0; k < DIM; k += 4) {
        v2f a, b;
        const float2 av = *(const float2*)(arow + k);   // contiguous b64 load
        a.x = av.x; a.y = av.y;
        const float* bp = bcol + (size_t)k * FEAT;
        b.x = bp[0];
        b.y = bp[FEAT];
        c = __builtin_amdgcn_wmma_f32_16x16x4_f32(
                /*neg_a=*/false, a, /*neg_b=*/false, b,
                /*c_mod=*/(short)0, c, /*reuse_a=*/false, /*reuse_b=*/false);
    }

    // C/D layout: lanes 0-15 hold M = 0..7 (vgpr r -> M=r), lanes 16-31 M = 8..15
    const int mbase = m0 + half * 8;
    const int col   = n0 + l;
    #pragma unroll
    for (int r = 0; r < 8; ++r)
        xp[(size_t)(mbase + r) * FEAT + col] = c[r];
}

// ---- 2: per-edge attention scores + segment max ----
// One wave per edge; lane indexes the U dimension; shfl-xor reduction per head.
__global__ __launch_bounds__(256) void gat_edge_scores(
        const float* __restrict__ xp, const int* __restrict__ edges,
        const float* __restrict__ ka1, const float* __restrict__ ba,
        float* __restrict__ scores, unsigned* __restrict__ smax) {
    const int wave = (blockIdx.x * blockDim.x + threadIdx.x) >> 5;
    const int lane = threadIdx.x & 31;
    if (wave >= N_EDGES) return;
    const int s = edges[wave * 2 + 0];
    const int t = edges[wave * 2 + 1];
    const float* xs = xp + (size_t)s * FEAT;
    const float* xt = xp + (size_t)t * FEAT;
    #pragma unroll
    for (int h = 0; h < HEADS; ++h) {
        const int idx = h * UNITS + lane;
        float f  = xs[idx] + xt[idx] + 2.0f * ba[idx];
        float lr = (f > 0.0f) ? f : SLOPE * f;
        float v  = lr * ka1[idx];
        #pragma unroll
        for (int off = 16; off > 0; off >>= 1)
            v += __shfl_xor(v, off, 32);
        if (lane == 0) {
            scores[(size_t)wave * HEADS + h] = v;
            atomicMax(&smax[(size_t)t * HEADS + h], enc_f(v));
        }
    }
}

// ---- 3: per-edge scatter of unnormalized softmax-weighted features ----
__global__ __launch_bounds__(256) void gat_edge_scatter(
        const float* __restrict__ xp, const int* __restrict__ edges,
        const float* __restrict__ scores, const unsigned* __restrict__ smax,
        float* __restrict__ den, float* __restrict__ out) {
    const int wave = (blockIdx.x * blockDim.x + threadIdx.x) >> 5;
    const int lane = threadIdx.x & 31;
    if (wave >= N_EDGES) return;
    const int s = edges[wave * 2 + 0];
    const int t = edges[wave * 2 + 1];
    const float* xs = xp + (size_t)s * FEAT;
    float* ot = out + (size_t)t * FEAT;

    float wv[HEADS];
    #pragma unroll
    for (int h = 0; h < HEADS; ++h) {
        float sc = scores[(size_t)wave * HEADS + h];
        float mx = dec_f(smax[(size_t)t * HEADS + h]);
        wv[h] = __expf(sc - mx);
    }
    if (lane < HEADS) atomicAdd(&den[(size_t)t * HEADS + lane], wv[lane]);
    #pragma unroll
    for (int h = 0; h < HEADS; ++h) {
        const int idx = h * UNITS + lane;
        atomicAdd(&ot[idx], wv[h] * xs[idx]);
    }
}

// ---- 4: finalize: out = elu(num/(den+eps) + bias) ----
__global__ void gat_finalize(float* __restrict__ out, const float* __restrict__ den,
                             const float* __restrict__ bias) {
    int i = blockIdx.x * blockDim.x + threadIdx.x;
    if (i >= N_NODES * FEAT) return;
    int n = i / FEAT, c = i % FEAT, h = c / UNITS;
    float v = out[i] / (den[n * HEADS + h] + EPS_F) + bias[c];
    out[i] = (v > 0.0f) ? v : (__expf(v) - 1.0f);
}

extern "C" void kernel_launch(void* const* d_in, const int* in_sizes, int n_in,
                              void* d_out, int out_size, void* d_ws, size_t ws_size,
                              hipStream_t stream) {
    const float* x    = (const float*)d_in[0];   // (N, 128)
    const int*   edg  = (const int*)  d_in[1];   // (E, 2) [src, tgt]
    const float* w    = (const float*)d_in[2];   // (128, 256)
    const float* ka1  = (const float*)d_in[3];   // (1, 8, 32) -> 256
    const float* ba   = (const float*)d_in[4];   // (1, 8, 32) -> 256
    const float* bias = (const float*)d_in[5];   // (256,)
    float* out = (float*)d_out;                  // (N, 256) f32

    // workspace layout (floats): xp | smax(enc u32) | den | scores  (~80 MB)
    float*    xp     = (float*)d_ws;
    unsigned* smax   = (unsigned*)(xp + (size_t)N_NODES * FEAT);
    float*    den    = (float*)(smax + (size_t)N_NODES * HEADS);
    float*    scores = den + (size_t)N_NODES * HEADS;

    gat_init<<<(N_NODES * FEAT + 255) / 256, 256, 0, stream>>>(out, smax, den);

    // 50000/16 * 16 = 50000 waves, 8 waves per 256-thread block
    gat_gemm_wmma<<<(N_NODES / 16) * 16 / 8, 256, 0, stream>>>(x, w, xp);

    gat_edge_scores<<<(N_EDGES + 7) / 8, 256, 0, stream>>>(xp, edg, ka1, ba, scores, smax);
    gat_edge_scatter<<<(N_EDGES + 7) / 8, 256, 0, stream>>>(xp, edg, scores, smax, den, out);
    gat_finalize<<<(N_NODES * FEAT + 255) / 256, 256, 0, stream>>>(out, den, bias);
}